// GNN_87866440941609
// MI455X (gfx1250) — compile-verified
//
#include <hip/hip_runtime.h>
#include <cmath>

// SAGEConv x2 + log_softmax for N=50000, E=800000, D_IN=64, D_HID=128, D_OUT=64
// Workspace layout (floats): sum1[N*64] | sum2[N*128] | deg[N] | h1[N*128]
// -> ~64.2 MB required in d_ws.

#define NN 50000
#define NE 800000
#define DIN 64
#define DHID 128
#define DOUT 64

typedef __attribute__((ext_vector_type(2))) float v2f;
typedef __attribute__((ext_vector_type(8))) float v8f;

__global__ void zero_kernel(float* __restrict__ p, int n) {
    int i = blockIdx.x * blockDim.x + threadIdx.x;
    if (i < n) p[i] = 0.0f;
}

// One thread per (edge, feature). f==0 also bumps the degree counter.
__global__ void scatter1_kernel(const float* __restrict__ x,
                                const int* __restrict__ ei,
                                float* __restrict__ sum1,
                                float* __restrict__ deg) {
    int tid = blockIdx.x * blockDim.x + threadIdx.x;
    if (tid >= NE * DIN) return;
    int e = tid >> 6;
    int f = tid & 63;
    int src = ei[e];
    int dst = ei[NE + e];
    atomicAdd(&sum1[dst * DIN + f], x[src * DIN + f]);
    if (f == 0) atomicAdd(&deg[dst], 1.0f);
}

__global__ void scatter2_kernel(const float* __restrict__ h1,
                                const int* __restrict__ ei,
                                float* __restrict__ sum2) {
    int tid = blockIdx.x * blockDim.x + threadIdx.x;
    if (tid >= NE * DHID) return;
    int e = tid >> 7;
    int f = tid & 127;
    int src = ei[e];
    int dst = ei[NE + e];
    atomicAdd(&sum2[dst * DHID + f], h1[src * DHID + f]);
}

// Layer 1: h1 = tanh(agg @ Wl^T + bl + x @ Wr^T), agg = sum1/max(deg,1)
// One wave (32 threads) per 16-node row tile. 50000/16 = 3125 tiles exactly.
__global__ __launch_bounds__(32) void gemm1_kernel(
    const float* __restrict__ x, const float* __restrict__ sum1,
    const float* __restrict__ deg,
    const float* __restrict__ Wl, const float* __restrict__ bl,
    const float* __restrict__ Wr, float* __restrict__ h1) {
    const int lane = threadIdx.x;          // 0..31, wave32
    const int m0   = blockIdx.x * 16;
    const int mr   = lane & 15;            // M row (A frags) / N col (B frags)
    const int kb   = (lane >> 4) * 2;      // K sub-offset within 4-wide step
    const int m    = m0 + mr;
    const float dinv = 1.0f / fmaxf(deg[m], 1.0f);

    v8f acc[8] = {};                       // 8 column tiles of 16 -> DHID=128

    // Path 1: mean-aggregated neighbors @ Wl^T
    for (int kk = 0; kk < DIN / 4; ++kk) {
        const int k = kk * 4 + kb;
        v2f a;
        a.x = sum1[m * DIN + k] * dinv;
        a.y = sum1[m * DIN + k + 1] * dinv;
#pragma unroll
        for (int nt = 0; nt < 8; ++nt) {
            const int n = nt * 16 + mr;
            v2f b;
            b.x = Wl[n * DIN + k];
            b.y = Wl[n * DIN + k + 1];
            acc[nt] = __builtin_amdgcn_wmma_f32_16x16x4_f32(
                false, a, false, b, (short)0, acc[nt], false, false);
        }
    }
    // Path 2: root features @ Wr^T
    for (int kk = 0; kk < DIN / 4; ++kk) {
        const int k = kk * 4 + kb;
        v2f a;
        a.x = x[m * DIN + k];
        a.y = x[m * DIN + k + 1];
#pragma unroll
        for (int nt = 0; nt < 8; ++nt) {
            const int n = nt * 16 + mr;
            v2f b;
            b.x = Wr[n * DIN + k];
            b.y = Wr[n * DIN + k + 1];
            acc[nt] = __builtin_amdgcn_wmma_f32_16x16x4_f32(
                false, a, false, b, (short)0, acc[nt], false, false);
        }
    }

    // Epilogue: bias + tanh. C/D layout: vgpr v -> row v + 8*(lane>>4).
    const int mo = m0 + 8 * (lane >> 4);
#pragma unroll
    for (int nt = 0; nt < 8; ++nt) {
        const int n = nt * 16 + mr;
        const float bias = bl[n];
#pragma unroll
        for (int v = 0; v < 8; ++v) {
            h1[(mo + v) * DHID + n] = tanhf(acc[nt][v] + bias);
        }
    }
}

// Layer 2: z = agg2 @ Wl^T + bl + h1 @ Wr^T, then row-wise log_softmax.
__global__ __launch_bounds__(32) void gemm2_kernel(
    const float* __restrict__ h1, const float* __restrict__ sum2,
    const float* __restrict__ deg,
    const float* __restrict__ Wl, const float* __restrict__ bl,
    const float* __restrict__ Wr, float* __restrict__ out) {
    __shared__ float zrow[16 * DOUT];
    const int lane = threadIdx.x;
    const int m0   = blockIdx.x * 16;
    const int mr   = lane & 15;
    const int kb   = (lane >> 4) * 2;
    const int m    = m0 + mr;
    const float dinv = 1.0f / fmaxf(deg[m], 1.0f);

    v8f acc[4] = {};                       // 4 column tiles -> DOUT=64

    for (int kk = 0; kk < DHID / 4; ++kk) {
        const int k = kk * 4 + kb;
        v2f a;
        a.x = sum2[m * DHID + k] * dinv;
        a.y = sum2[m * DHID + k + 1] * dinv;
#pragma unroll
        for (int nt = 0; nt < 4; ++nt) {
            const int n = nt * 16 + mr;
            v2f b;
            b.x = Wl[n * DHID + k];
            b.y = Wl[n * DHID + k + 1];
            acc[nt] = __builtin_amdgcn_wmma_f32_16x16x4_f32(
                false, a, false, b, (short)0, acc[nt], false, false);
        }
    }
    for (int kk = 0; kk < DHID / 4; ++kk) {
        const int k = kk * 4 + kb;
        v2f a;
        a.x = h1[m * DHID + k];
        a.y = h1[m * DHID + k + 1];
#pragma unroll
        for (int nt = 0; nt < 4; ++nt) {
            const int n = nt * 16 + mr;
            v2f b;
            b.x = Wr[n * DHID + k];
            b.y = Wr[n * DHID + k + 1];
            acc[nt] = __builtin_amdgcn_wmma_f32_16x16x4_f32(
                false, a, false, b, (short)0, acc[nt], false, false);
        }
    }

    // Stage z tile (16x64) in LDS for row-wise log_softmax.
    const int rbase = 8 * (lane >> 4);
#pragma unroll
    for (int nt = 0; nt < 4; ++nt) {
        const int n = nt * 16 + mr;
        const float bias = bl[n];
#pragma unroll
        for (int v = 0; v < 8; ++v) {
            zrow[(rbase + v) * DOUT + n] = acc[nt][v] + bias;
        }
    }
    __syncthreads();  // single-wave block: LDS wait only

    if (lane < 16) {
        const float* zr = &zrow[lane * DOUT];
        float mx = zr[0];
        for (int c = 1; c < DOUT; ++c) mx = fmaxf(mx, zr[c]);
        float s = 0.0f;
        for (int c = 0; c < DOUT; ++c) s += expf(zr[c] - mx);
        const float lse = mx + logf(s);
        float* orow = &out[(m0 + lane) * DOUT];
        for (int c = 0; c < DOUT; ++c) orow[c] = zr[c] - lse;
    }
}

extern "C" void kernel_launch(void* const* d_in, const int* in_sizes, int n_in,
                              void* d_out, int out_size, void* d_ws, size_t ws_size,
                              hipStream_t stream) {
    const float* x   = (const float*)d_in[0];
    const int*   ei  = (const int*)d_in[1];   // edge_index [2, E]
    const float* Wl1 = (const float*)d_in[2];
    const float* bl1 = (const float*)d_in[3];
    const float* Wr1 = (const float*)d_in[4];
    const float* Wl2 = (const float*)d_in[5];
    const float* bl2 = (const float*)d_in[6];
    const float* Wr2 = (const float*)d_in[7];
    float* out = (float*)d_out;

    float* ws   = (float*)d_ws;
    float* sum1 = ws;                          // N*64
    float* sum2 = sum1 + (size_t)NN * DIN;     // N*128
    float* degb = sum2 + (size_t)NN * DHID;    // N
    float* h1   = degb + NN;                   // N*128 (no zeroing needed)

    const int nz = NN * (DIN + DHID) + NN;     // sum1 + sum2 + deg
    zero_kernel<<<(nz + 255) / 256, 256, 0, stream>>>(ws, nz);

    scatter1_kernel<<<(NE * DIN + 255) / 256, 256, 0, stream>>>(x, ei, sum1, degb);
    gemm1_kernel<<<NN / 16, 32, 0, stream>>>(x, sum1, degb, Wl1, bl1, Wr1, h1);
    scatter2_kernel<<<(NE * DHID + 255) / 256, 256, 0, stream>>>(h1, ei, sum2);
    gemm2_kernel<<<NN / 16, 32, 0, stream>>>(h1, sum2, degb, Wl2, bl2, Wr2, out);
}